// GraphNorm_7069516169366
// MI455X (gfx1250) — compile-verified
//
#include <hip/hip_runtime.h>
#include <hip/hip_bf16.h>
#include <math.h>

// GraphNorm on MI455X (gfx1250), compile-only loop.
// Grid: one workgroup per (graph, 128-channel slice) -> B*4 = 512 independent
// WGs of 512 threads (16 waves). Channel slices are fully independent, so no
// cross-block communication is needed and concurrency is 4x the graph count.
//   Pass 1: async global->LDS (ASYNCcnt path), 4-deep pipelined 32-node tiles
//           (48 KB in flight per WG -> ~24 MB machine-wide, above the
//           23.3 TB/s x latency product), accumulate sum(x), sum(x^2).
//   Stats : var = E[x^2] - 2*mm*E[x] + mm^2, mm = mean*msc (single-sweep fusion)
//   Pass 2: re-read x (L2-resident: 128MB x < 192MB L2), reg-held scale/bias,
//           non-temporal load/store so `out` never evicts x from L2.

#define HID 512
#define CH 128                         // channels per block slice
#define SLICES (HID / CH)              // 4 blocks per graph
#define NTHREADS 512
#define TILE_NODES 32
#define TILE_ELEMS (TILE_NODES * CH)   // 4096 floats = 16 KB per tile
#define ISSUES_PER_TILE (TILE_ELEMS / (NTHREADS * 4)) // 2 x B128 per thread
#define NBUF 4                         // async pipeline depth

#if defined(__HIP_DEVICE_COMPILE__) && __has_builtin(__builtin_amdgcn_global_load_async_to_lds_b128)
#define HAVE_ASYNC_LDS 1
#else
#define HAVE_ASYNC_LDS 0
#endif

typedef float v4f __attribute__((ext_vector_type(4)));
typedef int   v4i __attribute__((ext_vector_type(4)));

#if HAVE_ASYNC_LDS
typedef __attribute__((address_space(1))) v4i* gv4i_t; // global int4*
typedef __attribute__((address_space(3))) v4i* lv4i_t; // LDS int4*
#endif

__device__ __forceinline__ void async_copy_b128(const float* gsrc, float* ldst) {
#if HAVE_ASYNC_LDS
  // Generic-LDS addr: low 32 bits are the LDS byte offset (flat aperture
  // truncation rule), so an integer round-trip yields a valid as(3) pointer.
  __builtin_amdgcn_global_load_async_to_lds_b128(
      (gv4i_t)(unsigned long long)gsrc,
      (lv4i_t)(unsigned)(unsigned long long)ldst,
      /*offset=*/0, /*cpol=*/0);
#else
  (void)gsrc; (void)ldst;
#endif
}

template <int N>
__device__ __forceinline__ void wait_asynccnt() {
#if defined(__HIP_DEVICE_COMPILE__)
#if __has_builtin(__builtin_amdgcn_s_wait_asynccnt)
  __builtin_amdgcn_s_wait_asynccnt(N);
#else
  asm volatile("s_wait_asynccnt %0" ::"n"(N) : "memory");
#endif
#endif
}

__global__ __launch_bounds__(NTHREADS) void graphnorm512_kernel(
    const float* __restrict__ x, const float* __restrict__ gnw,
    const float* __restrict__ gnb, const float* __restrict__ msc,
    const long long* __restrict__ batch_num, float* __restrict__ out) {
#if HAVE_ASYNC_LDS
  __shared__ __align__(16) float stage[NBUF][TILE_ELEMS]; // 64 KB, 4-deep
#endif
  __shared__ float psum[4][CH];   // 2 KB
  __shared__ float psum2[4][CH];  // 2 KB
  __shared__ __align__(16) float s_scale[CH]; // 512 B
  __shared__ __align__(16) float s_bias[CH];  // 512 B
  __shared__ long long s_start;
  __shared__ int s_n;

  const int tid = threadIdx.x;
  const int b  = blockIdx.x >> 2;            // graph id
  const int cg = blockIdx.x & (SLICES - 1);  // channel slice 0..3

  if (tid == 0) {
    long long st = 0;
    for (int i = 0; i < b; ++i) st += batch_num[i];
    s_start = st;
    s_n = (int)batch_num[b];
  }
  __syncthreads();

  const long long start = s_start;
  const int n = s_n;                   // nodes in this graph (512 in harness)
  const int c = tid & (CH - 1);        // channel (within slice) for reduction
  const int h = tid >> 7;              // node-quarter 0..3
  const float* __restrict__ gx = x + (start << 9) + cg * CH;
  float* __restrict__ gox = out + (start << 9) + cg * CH;

  float s0 = 0.f, s1 = 0.f;

#if HAVE_ASYNC_LDS
  const int nt = (n + TILE_NODES - 1) / TILE_NODES;
  const int rmax = n - 1;

  // Rows are CLAMPED, never EXEC-masked, so every wave issues exactly
  // ISSUES_PER_TILE async instructions per tile -> exact ASYNCcnt math.
  // Tail nodes are masked only in the accumulate.
  auto issue = [&](int t) {
#pragma unroll
    for (int k = 0; k < ISSUES_PER_TILE; ++k) {
      int e = k * (NTHREADS * 4) + tid * 4;  // element within tile
      int row = t * TILE_NODES + (e >> 7);   // 128 floats per row
      if (row > rmax) row = rmax;
      async_copy_b128(gx + (long long)row * HID + (e & (CH - 1)),
                      &stage[t & (NBUF - 1)][e]);
    }
  };

  for (int p = 0; p < NBUF - 1 && p < nt; ++p) issue(p); // prime the pipeline
  for (int t = 0; t < nt; ++t) {
    if (t + (NBUF - 1) < nt) issue(t + (NBUF - 1));      // uniform guard
    // Issued tiles = min(nt, t+NBUF); need tile t complete ->
    // allowed outstanding = ISSUES_PER_TILE * min(NBUF-1, nt-1-t).
    int rem = nt - 1 - t; if (rem > NBUF - 1) rem = NBUF - 1;
    switch (rem) {                       // s_wait_asynccnt takes SIMM16 only
      case 3:  wait_asynccnt<3 * ISSUES_PER_TILE>(); break;
      case 2:  wait_asynccnt<2 * ISSUES_PER_TILE>(); break;
      case 1:  wait_asynccnt<1 * ISSUES_PER_TILE>(); break;
      default: wait_asynccnt<0>(); break;
    }
    __syncthreads();                     // everyone's tile-t loads complete
    const float* sb = stage[t & (NBUF - 1)];
#pragma unroll
    for (int j = 0; j < 8; ++j) {
      int node = t * TILE_NODES + h * 8 + j;
      float v = sb[(h * 8 + j) * CH + c];
      if (node < n) { s0 += v; s1 += v * v; }
    }
    __syncthreads();   // buffer consumed; safe to refill at iteration t+1
  }
#else
  // Fallback: direct coalesced accumulation (no async builtins available).
  for (int r = h; r < n; r += 4) {
    float v = gx[(long long)r * HID + c];
    s0 += v; s1 += v * v;
  }
#endif

  psum[h][c] = s0;
  psum2[h][c] = s1;
  __syncthreads();

  if (tid < CH) {
    float tot  = psum[0][tid] + psum[1][tid] + psum[2][tid] + psum[3][tid];
    float tot2 = psum2[0][tid] + psum2[1][tid] + psum2[2][tid] + psum2[3][tid];
    const int ch = cg * CH + tid;      // global channel index
    float invn = 1.0f / (float)n;
    float m   = tot * invn;            // E[x]
    float ex2 = tot2 * invn;           // E[x^2]
    float mm  = m * msc[ch];           // mean*msc
    float var = ex2 - 2.0f * mm * m + mm * mm; // E[(x-mm)^2]
    float sc  = gnw[ch] * (1.0f / sqrtf(var + 1e-6f));
    s_scale[tid] = sc;
    s_bias[tid]  = gnb[ch] - mm * sc;
  }
  __syncthreads();

  // Pass 2: each thread's channel quad is FIXED -> scale/bias live in regs;
  // inner loop is pure NT load -> FMA -> NT store, x re-read hits L2.
  const int c0 = (tid * 4) & (CH - 1);        // constant column quad
  const int r0 = tid >> 5;                    // starting row 0..15
  const v4f sc4 = *(const v4f*)&s_scale[c0];
  const v4f bi4 = *(const v4f*)&s_bias[c0];
  for (int row = r0; row < n; row += (NTHREADS * 4) / CH) {
    const float* xp = gx + (long long)row * HID + c0;
    float* op = gox + (long long)row * HID + c0;
    v4f xx = __builtin_nontemporal_load((const v4f*)xp);
    v4f r = xx * sc4 + bi4;
    __builtin_nontemporal_store(r, (v4f*)op);
  }
}

extern "C" void kernel_launch(void* const* d_in, const int* in_sizes, int n_in,
                              void* d_out, int out_size, void* d_ws, size_t ws_size,
                              hipStream_t stream) {
  const float* x        = (const float*)d_in[0];
  const float* gnw      = (const float*)d_in[1];
  const float* gnb      = (const float*)d_in[2];
  const float* msc      = (const float*)d_in[3];
  const long long* bn   = (const long long*)d_in[4];
  float* out            = (float*)d_out;

  const int B = in_sizes[4]; // number of graphs (128)
  graphnorm512_kernel<<<dim3(B * SLICES), dim3(NTHREADS), 0, stream>>>(
      x, gnw, gnb, msc, bn, out);

  (void)n_in; (void)out_size; (void)d_ws; (void)ws_size;
}